// QuantumHeadAmplitude_54692113547573
// MI455X (gfx1250) — compile-verified
//
#include <hip/hip_runtime.h>
#include <cstdint>

// ---------------------------------------------------------------------------
// QuantumHeadAmplitude on MI455X (gfx1250, wave32, WMMA f32 16x16x4)
//
// State: 256 batches x 2^16 complex64 amplitudes = 128 MB, kept in-place in
// d_ws (L2-resident on MI455X: 192 MB L2). Each layer's 16 Rot gates are
// Kron-fused into four 16x16 complex unitaries (one per 4-qubit group) and
// applied with V_WMMA_F32_16X16X4_F32. CNOT rings are XOR-linear basis
// permutations folded into address generation (per-bit XOR tables) -> no
// data-movement passes for entanglement layers.
// ---------------------------------------------------------------------------

typedef __attribute__((ext_vector_type(2))) float v2f;
typedef __attribute__((ext_vector_type(8))) float v8f;

#define QH_BATCH 256
#define QH_DIM   65536          // 2^16 amplitudes
#define QH_TPW   8              // tiles (16x16) per wave in a gate pass

// ---- workspace layout (bytes) ----
#define WS_STATE_OFF 0                                   // float2[256*65536] = 128 MB
#define WS_GATES_OFF (size_t)(QH_BATCH * (size_t)QH_DIM * 8)  // float[2][4][2][256]
#define WS_INVN_OFF  (WS_GATES_OFF + 4096 * 4)           // float[256]
#define WS_BM_OFF    (WS_INVN_OFF + 256 * 4)             // uint32[48]: id, Pinv(l0), R∘P fwd

// ---------------------------------------------------------------------------
// WMMA helper: D = A(16x4) * B(4x16) + C  (f32, wave32)
// ---------------------------------------------------------------------------
__device__ __forceinline__ v8f wmma4(v2f a, v2f b, v8f c) {
  // 8 args: (neg_a, A, neg_b, B, c_mod, C, reuse_a, reuse_b); f32 NEG bits must be 0.
  return __builtin_amdgcn_wmma_f32_16x16x4_f32(false, a, false, b, (short)0, c,
                                               false, false);
}

// ---------------------------------------------------------------------------
// CNOT ring permutations on 16-bit basis indices (wire w == bit 15-w).
// Both forward and inverse cascades are XOR-linear maps over GF(2).
// ---------------------------------------------------------------------------
__device__ inline uint32_t cnot_fwd(uint32_t i, int r) {
  for (int w = 0; w < 16; ++w) {
    int c = 15 - w;
    int t = 15 - ((w + r) & 15);
    i ^= ((i >> c) & 1u) << t;
  }
  return i;
}
__device__ inline uint32_t cnot_inv(uint32_t i, int r) {
  for (int w = 15; w >= 0; --w) {
    int c = 15 - w;
    int t = 15 - ((w + r) & 15);
    i ^= ((i >> c) & 1u) << t;
  }
  return i;
}

__device__ __forceinline__ uint32_t xmap16(uint32_t x, const uint32_t bmr[16]) {
  uint32_t r = 0;
#pragma unroll
  for (int i = 0; i < 16; ++i) r ^= ((x >> i) & 1u) ? bmr[i] : 0u;
  return r;
}

// ---------------------------------------------------------------------------
// Setup: build 8 fused 16x16 complex unitaries (Kron of 4 Rot gates) and the
// three per-bit XOR address-map tables.
// gates layout: [l][g][re/im][row*16+col], 512 floats per (l,g).
// ---------------------------------------------------------------------------
__global__ void qh_setup(const float* __restrict__ qp, float* __restrict__ gates,
                         uint32_t* __restrict__ bms) {
  const int t = threadIdx.x;       // 0..255 -> (row, col) of the 16x16 matrix
  const int r = t >> 4;
  const int c = t & 15;
  for (int l = 0; l < 2; ++l) {
    for (int g = 0; g < 4; ++g) {
      float pr = 1.f, pi = 0.f;
      for (int q = 0; q < 4; ++q) {            // q=0 is MSB of the group index
        const int w = 4 * g + q;
        const float phi = qp[(l * 16 + w) * 3 + 0];
        const float th  = qp[(l * 16 + w) * 3 + 1];
        const float om  = qp[(l * 16 + w) * 3 + 2];
        const float ch = cosf(0.5f * th), sh = sinf(0.5f * th);
        const float a  = 0.5f * (phi + om), bb = 0.5f * (phi - om);
        const int br = (r >> (3 - q)) & 1;
        const int bc = (c >> (3 - q)) & 1;
        float er, ei;
        if (br == 0 && bc == 0)      { er =  cosf(a)  * ch; ei = -sinf(a)  * ch; }
        else if (br == 0 && bc == 1) { er = -cosf(bb) * sh; ei = -sinf(bb) * sh; }
        else if (br == 1 && bc == 0) { er =  cosf(bb) * sh; ei = -sinf(bb) * sh; }
        else                         { er =  cosf(a)  * ch; ei =  sinf(a)  * ch; }
        const float nr = pr * er - pi * ei;
        const float ni = pr * ei + pi * er;
        pr = nr; pi = ni;
      }
      gates[((l * 4 + g) * 2 + 0) * 256 + t] = pr;
      gates[((l * 4 + g) * 2 + 1) * 256 + t] = pi;
    }
  }
  if (t < 16) {
    bms[t]      = 1u << t;                                  // identity
    bms[16 + t] = cnot_inv(1u << t, 1);                     // Pinv of layer-0 ring
    bms[32 + t] = cnot_fwd(cnot_fwd(1u << t, 1), 2);        // R(P(.)) forward compose
  }
}

// ---------------------------------------------------------------------------
// Per-batch inverse L2 norm of x.
// ---------------------------------------------------------------------------
__global__ void qh_norm(const float* __restrict__ x, float* __restrict__ invn) {
  __shared__ float red[256];
  const int b = blockIdx.x, tid = threadIdx.x;
  const float* xb = x + ((size_t)b << 16);
  float s = 0.f;
  for (int it = 0; it < 256; ++it) {
    const float v = xb[it * 256 + tid];
    s += v * v;
  }
  red[tid] = s;
  __syncthreads();
  for (int k = 128; k > 0; k >>= 1) {
    if (tid < k) red[tid] += red[tid + k];
    __syncthreads();
  }
  if (tid == 0) invn[b] = rsqrtf(red[0]);
}

// ---------------------------------------------------------------------------
// First gate pass: layer 0, group 0 (wires 0-3, field bits 15..12), reading
// the REAL input x (normalized) -> complex state. Identity address map.
// Out_r = Ur*S, Out_i = Ui*S : 8 WMMAs per tile.
// ---------------------------------------------------------------------------
__global__ void qh_pass0(const float* __restrict__ x, const float* __restrict__ invn,
                         float2* __restrict__ st, const float* __restrict__ g512) {
  const int lane = threadIdx.x & 31;
  const int h = lane >> 4;          // wave half
  const int n = lane & 15;          // B/D column, A row
  const float* gr = g512;
  const float* gi = g512 + 256;

  v2f Ar[4], Ai[4];
#pragma unroll
  for (int kk = 0; kk < 4; ++kk) {
    const int k0 = 4 * kk + 2 * h;
    Ar[kk][0] = gr[n * 16 + k0]; Ar[kk][1] = gr[n * 16 + k0 + 1];
    Ai[kk][0] = gi[n * 16 + k0]; Ai[kk][1] = gi[n * 16 + k0 + 1];
  }

  const int gw = blockIdx.x * 8 + (threadIdx.x >> 5);
  for (int tt = 0; tt < QH_TPW; ++tt) {
    const int tile = gw * QH_TPW + tt;
    const int b = tile >> 8;
    const uint32_t o = (uint32_t)(tile & 255);
    const uint32_t C0 = (o << 4) | (uint32_t)n;    // outer bits 11..4, col bits 3..0
    const float* xb = x + ((size_t)b << 16);
    float2* stb = st + ((size_t)b << 16);
    const float sc = invn[b];

    v8f aor = {0.f, 0.f, 0.f, 0.f, 0.f, 0.f, 0.f, 0.f};
    v8f aoi = {0.f, 0.f, 0.f, 0.f, 0.f, 0.f, 0.f, 0.f};
#pragma unroll
    for (int kk = 0; kk < 4; ++kk) {
      const uint32_t k0 = (uint32_t)(4 * kk + 2 * h);
      v2f Br;
      Br[0] = xb[C0 | (k0 << 12)] * sc;
      Br[1] = xb[C0 | ((k0 + 1) << 12)] * sc;
      aor = wmma4(Ar[kk], Br, aor);
      aoi = wmma4(Ai[kk], Br, aoi);
    }
#pragma unroll
    for (int v = 0; v < 8; ++v) {
      const uint32_t j = (uint32_t)(v + 8 * h);
      stb[C0 | (j << 12)] = make_float2(aor[v], aoi[v]);
    }
  }
}

// ---------------------------------------------------------------------------
// Generic in-place complex gate pass. The 4-bit contracted field sits at
// shift `cs`, the 4-bit tile-column field at shift `ns`, the remaining 8 bits
// come from the outer tile index o via base16 = ((o&hiM)<<hiS)|((o&loM)<<loS).
// All addresses are pushed through the XOR-linear per-bit map `bm` (identity
// for layer 0; layer-0-CNOT-inverse for layer 1 -> entanglement layer costs
// nothing). Reads and writes hit identical addresses -> in-place safe.
// ---------------------------------------------------------------------------
__global__ void qh_pass(float2* __restrict__ st, const float* __restrict__ g512,
                        const uint32_t* __restrict__ bm, int cs, int ns,
                        int hiM, int hiS, int loM, int loS) {
  const int lane = threadIdx.x & 31;
  const int h = lane >> 4;
  const int n = lane & 15;
  const float* gr = g512;
  const float* gi = g512 + 256;

  uint32_t bmr[16];
#pragma unroll
  for (int i = 0; i < 16; ++i) bmr[i] = bm[i];
  const uint32_t t0 = bm[cs + 0], t1 = bm[cs + 1], t2 = bm[cs + 2], t3 = bm[cs + 3];
  const uint32_t u0 = bm[ns + 0], u1 = bm[ns + 1], u2 = bm[ns + 2], u3 = bm[ns + 3];
  const uint32_t cn = ((n & 1) ? u0 : 0u) ^ ((n & 2) ? u1 : 0u) ^
                      ((n & 4) ? u2 : 0u) ^ ((n & 8) ? u3 : 0u);

  v2f Ar[4], Ai[4], An[4];
#pragma unroll
  for (int kk = 0; kk < 4; ++kk) {
    const int k0 = 4 * kk + 2 * h;
    Ar[kk][0] = gr[n * 16 + k0]; Ar[kk][1] = gr[n * 16 + k0 + 1];
    Ai[kk][0] = gi[n * 16 + k0]; Ai[kk][1] = gi[n * 16 + k0 + 1];
    An[kk] = -Ai[kk];
  }

  const int gw = blockIdx.x * 8 + (threadIdx.x >> 5);
  for (int tt = 0; tt < QH_TPW; ++tt) {
    const int tile = gw * QH_TPW + tt;
    const int b = tile >> 8;
    const uint32_t o = (uint32_t)(tile & 255);
    const uint32_t base16 =
        ((o & (uint32_t)hiM) << hiS) | ((o & (uint32_t)loM) << loS);
    const uint32_t C0 = xmap16(base16, bmr) ^ cn;
    float2* stb = st + ((size_t)b << 16);

    v8f aor = {0.f, 0.f, 0.f, 0.f, 0.f, 0.f, 0.f, 0.f};
    v8f aoi = {0.f, 0.f, 0.f, 0.f, 0.f, 0.f, 0.f, 0.f};
#pragma unroll
    for (int kk = 0; kk < 4; ++kk) {
      const uint32_t k0 = (uint32_t)(4 * kk + 2 * h);
      const uint32_t r0 = ((k0 & 2) ? t1 : 0u) ^ ((k0 & 4) ? t2 : 0u) ^
                          ((k0 & 8) ? t3 : 0u);           // k0 even -> no t0 term
      const uint32_t r1 = r0 ^ t0;
      const float2 e0 = stb[C0 ^ r0];
      const float2 e1 = stb[C0 ^ r1];
      v2f Br; Br[0] = e0.x; Br[1] = e1.x;
      v2f Bi; Bi[0] = e0.y; Bi[1] = e1.y;
      aor = wmma4(Ar[kk], Br, aor);   // + Ur*Sr
      aor = wmma4(An[kk], Bi, aor);   // - Ui*Si
      aoi = wmma4(Ar[kk], Bi, aoi);   // + Ur*Si
      aoi = wmma4(Ai[kk], Br, aoi);   // + Ui*Sr
    }
#pragma unroll
    for (int v = 0; v < 8; ++v) {
      const uint32_t j = (uint32_t)(v + 8 * h);
      const uint32_t rj = ((j & 1) ? t0 : 0u) ^ ((j & 2) ? t1 : 0u) ^
                          ((j & 4) ? t2 : 0u) ^ ((j & 8) ? t3 : 0u);
      stb[C0 ^ rj] = make_float2(aor[v], aoi[v]);
    }
  }
}

// ---------------------------------------------------------------------------
// Final pass: |amp|^2, per-wire Z expvals with sign bits taken from the fully
// composed CNOT permutation (R o P forward map applied to the stored index),
// then the 16->10 linear head. One block per batch element.
// ---------------------------------------------------------------------------
__global__ void qh_expval_head(const float2* __restrict__ st,
                               const uint32_t* __restrict__ bmRP,
                               const float* __restrict__ W,
                               const float* __restrict__ bias,
                               float* __restrict__ out) {
  __shared__ float red[256];
  __shared__ float fw[16];
  const int b = blockIdx.x, tid = threadIdx.x;
  uint32_t bmr[16];
#pragma unroll
  for (int i = 0; i < 16; ++i) bmr[i] = bmRP[i];
  const uint32_t jt = xmap16((uint32_t)tid, bmr);   // low 8 bits of the index
  const float2* stb = st + ((size_t)b << 16);

  float acc[16];
#pragma unroll
  for (int w = 0; w < 16; ++w) acc[w] = 0.f;

  for (int it = 0; it < 256; ++it) {
    const uint32_t ji = xmap16((uint32_t)it << 8, bmr) ^ jt;
    const float2 a = stb[it * 256 + tid];
    const float p = a.x * a.x + a.y * a.y;
#pragma unroll
    for (int w = 0; w < 16; ++w)
      acc[w] += ((ji >> (15 - w)) & 1u) ? -p : p;
  }

#pragma unroll
  for (int w = 0; w < 16; ++w) {
    red[tid] = acc[w];
    __syncthreads();
    for (int k = 128; k > 0; k >>= 1) {
      if (tid < k) red[tid] += red[tid + k];
      __syncthreads();
    }
    if (tid == 0) fw[w] = red[0];
    __syncthreads();
  }

  if (tid < 10) {
    float s = bias[tid];
#pragma unroll
    for (int w = 0; w < 16; ++w) s += fw[w] * W[tid * 16 + w];
    out[b * 10 + tid] = s;
  }
}

// ---------------------------------------------------------------------------
extern "C" void kernel_launch(void* const* d_in, const int* in_sizes, int n_in,
                              void* d_out, int out_size, void* d_ws, size_t ws_size,
                              hipStream_t stream) {
  (void)in_sizes; (void)n_in; (void)out_size; (void)ws_size;
  const float* x    = (const float*)d_in[0];   // (256, 65536)
  const float* qp   = (const float*)d_in[1];   // (2, 16, 3)
  const float* W    = (const float*)d_in[2];   // (10, 16)
  const float* bias = (const float*)d_in[3];   // (10,)
  float* out = (float*)d_out;                  // (256, 10)

  char* wsb = (char*)d_ws;
  float2*   state = (float2*)(wsb + WS_STATE_OFF);
  float*    gates = (float*)(wsb + WS_GATES_OFF);
  float*    invn  = (float*)(wsb + WS_INVN_OFF);
  uint32_t* bms   = (uint32_t*)(wsb + WS_BM_OFF);
  const uint32_t* bmId   = bms;
  const uint32_t* bmPinv = bms + 16;
  const uint32_t* bmRP   = bms + 32;

  qh_setup<<<1, 256, 0, stream>>>(qp, gates, bms);
  qh_norm<<<QH_BATCH, 256, 0, stream>>>(x, invn);

  const dim3 pg(1024);   // 1024 blocks * 8 waves * 8 tiles = 65536 tiles

  // Layer 0: groups 0..3 (group 0 consumes real x with normalization).
  qh_pass0<<<pg, 256, 0, stream>>>(x, invn, state, gates + 0 * 512);
  qh_pass<<<pg, 256, 0, stream>>>(state, gates + 1 * 512, bmId, 8, 0, 0xF0, 8, 0x0F, 4);
  qh_pass<<<pg, 256, 0, stream>>>(state, gates + 2 * 512, bmId, 4, 0, 0xFF, 8, 0x00, 0);
  qh_pass<<<pg, 256, 0, stream>>>(state, gates + 3 * 512, bmId, 0, 4, 0xFF, 8, 0x00, 0);
  // Layer 1: identical passes, addresses routed through layer-0 CNOT inverse.
  qh_pass<<<pg, 256, 0, stream>>>(state, gates + 4 * 512, bmPinv, 12, 0, 0xFF, 4, 0x00, 0);
  qh_pass<<<pg, 256, 0, stream>>>(state, gates + 5 * 512, bmPinv, 8, 0, 0xF0, 8, 0x0F, 4);
  qh_pass<<<pg, 256, 0, stream>>>(state, gates + 6 * 512, bmPinv, 4, 0, 0xFF, 8, 0x00, 0);
  qh_pass<<<pg, 256, 0, stream>>>(state, gates + 7 * 512, bmPinv, 0, 4, 0xFF, 8, 0x00, 0);

  qh_expval_head<<<QH_BATCH, 256, 0, stream>>>(state, bmRP, W, bias, out);
}